// GraphAttentionLayer_2869038154140
// MI455X (gfx1250) — compile-verified
//
#include <hip/hip_runtime.h>

typedef __attribute__((ext_vector_type(2))) float v2f;
typedef __attribute__((ext_vector_type(8))) float v8f;

#define N_NODES 8192
#define IN_FEAT 512
#define OUT_FEAT 256
#define ALPHA 0.2f
#define NEG_BIG (-9000000000000000.0f)

// ---------------------------------------------------------------------------
// Kernel 1: h = x @ W  (8192x512 @ 512x256), fp32 WMMA 16x16x4.
// One wave (32 lanes) computes a 16x16 tile of h; 128 chained WMMAs over K.
// ---------------------------------------------------------------------------
__global__ __launch_bounds__(32) void gemm_h_kernel(const float* __restrict__ x,
                                                    const float* __restrict__ W,
                                                    float* __restrict__ h) {
    const int lane = threadIdx.x;          // 0..31
    const int c0   = blockIdx.x * 16;      // output col tile
    const int r0   = blockIdx.y * 16;      // output row tile
    const int m16  = lane & 15;
    const int koff = (lane >> 4) << 1;     // 0 for lanes 0-15, 2 for lanes 16-31
    const int rsel = (lane >> 4) << 3;     // C rows: +0 / +8

    v8f acc = {};
    for (int k = 0; k < IN_FEAT; k += 4) {
        // A fragment: A[M=m16][K=k+koff], A[M=m16][K=k+koff+1]  (16x4 f32 layout)
        v2f a = *(const v2f*)(x + (size_t)(r0 + m16) * IN_FEAT + k + koff);
        // B fragment: B[K=k+koff][N=m16], B[K=k+koff+1][N=m16]  (4x16 f32 layout)
        v2f b;
        b.x = W[(size_t)(k + koff) * OUT_FEAT + c0 + m16];
        b.y = W[(size_t)(k + koff + 1) * OUT_FEAT + c0 + m16];
        acc = __builtin_amdgcn_wmma_f32_16x16x4_f32(false, a, false, b,
                                                    (short)0, acc, false, false);
    }
#pragma unroll
    for (int v = 0; v < 8; ++v)
        h[(size_t)(r0 + rsel + v) * OUT_FEAT + c0 + m16] = acc[v];
}

// ---------------------------------------------------------------------------
// Kernel 2: src = h @ a1, dst = h @ a2. One wave per row, shuffle reduction.
// ---------------------------------------------------------------------------
__global__ __launch_bounds__(256) void attn_coeff_kernel(const float* __restrict__ h,
                                                         const float* __restrict__ a,
                                                         float* __restrict__ src,
                                                         float* __restrict__ dst) {
    const int lane = threadIdx.x & 31;
    const int wave = threadIdx.x >> 5;
    const int row  = blockIdx.x * 8 + wave;
    float s = 0.f, d = 0.f;
    for (int c = lane; c < OUT_FEAT; c += 32) {
        float hv = h[(size_t)row * OUT_FEAT + c];
        s += hv * a[c];
        d += hv * a[OUT_FEAT + c];
    }
#pragma unroll
    for (int off = 16; off > 0; off >>= 1) {
        s += __shfl_xor(s, off, 32);
        d += __shfl_xor(d, off, 32);
    }
    if (lane == 0) { src[row] = s; dst[row] = d; }
}

// ---------------------------------------------------------------------------
// Kernel 3: fused masked-softmax + (att @ h) + leaky_relu, flash-style.
// Block = 256 threads (8 waves) owns 16 rows. Each wave owns a 16x32 slab of
// the fp32 accumulator (2 WMMA C tiles). Online softmax in LDS.
// ---------------------------------------------------------------------------
__global__ __launch_bounds__(256) void gat_attend_kernel(const float* __restrict__ h,
                                                         const int* __restrict__ adj,
                                                         const float* __restrict__ src,
                                                         const float* __restrict__ dst,
                                                         float* __restrict__ out) {
    __shared__ float sLogit[16][34];   // padded: keeps ds_load_b64 aligned, banks spread
    __shared__ float sP[16][34];
    __shared__ float sM[16], sMnew[16], sF[16], sL[16], sSrc[16];

    const int tid  = threadIdx.x;
    const int lane = tid & 31;
    const int wave = tid >> 5;
    const int r0   = blockIdx.x * 16;

    const int li = tid >> 4;           // 0..15 : row within tile (coalesced adj reads)
    const int lj = tid & 15;           // 0..15 : col base; second elem at lj+16

    if (tid < 16) {
        sM[tid]   = -INFINITY;
        sL[tid]   = 0.f;
        sSrc[tid] = src[r0 + tid];
    }
    __syncthreads();

    const int m16   = lane & 15;
    const int koff  = (lane >> 4) << 1;   // A/B fragment K split: 0 / 2
    const int rsel  = (lane >> 4) << 3;   // C row split: 0 / 8
    const int cbase = wave * 32;          // this wave's 32 output columns

    v8f acc0 = {};   // cols [cbase, cbase+16)
    v8f acc1 = {};   // cols [cbase+16, cbase+32)

    for (int j0 = 0; j0 < N_NODES; j0 += 32) {
        // --- 1: masked leaky-relu logits for this 16x32 tile -> LDS ---
        {
            const size_t arow = (size_t)(r0 + li) * N_NODES + j0;
            if (j0 + 32 < N_NODES)
                __builtin_prefetch(&adj[arow + 32 + lj], 0, 1);  // global_prefetch_b8
            const float si = sSrc[li];
            int av0 = adj[arow + lj];
            int av1 = adj[arow + lj + 16];
            float e0 = si + dst[j0 + lj];
            float e1 = si + dst[j0 + lj + 16];
            e0 = e0 > 0.f ? e0 : ALPHA * e0;
            e1 = e1 > 0.f ? e1 : ALPHA * e1;
            sLogit[li][lj]      = (av0 > 0) ? e0 : NEG_BIG;
            sLogit[li][lj + 16] = (av1 > 0) ? e1 : NEG_BIG;
        }
        __syncthreads();

        // --- 2: per-row online-softmax max update ---
        if (tid < 16) {
            float tm = sLogit[tid][0];
#pragma unroll
            for (int q = 1; q < 32; ++q) tm = fmaxf(tm, sLogit[tid][q]);
            float mOld = sM[tid];
            float mNew = fmaxf(mOld, tm);
            sMnew[tid] = mNew;
            sF[tid]    = __expf(mOld - mNew);   // 0 on first tile (mOld = -inf)
            sM[tid]    = mNew;
        }
        __syncthreads();

        // --- 3: exponentiate -> P tile ---
        {
            float mn = sMnew[li];
            sP[li][lj]      = __expf(sLogit[li][lj]      - mn);
            sP[li][lj + 16] = __expf(sLogit[li][lj + 16] - mn);
        }
        __syncthreads();

        // --- 4: running-sum update ---
        if (tid < 16) {
            float s = 0.f;
#pragma unroll
            for (int q = 0; q < 32; ++q) s += sP[tid][q];
            sL[tid] = sL[tid] * sF[tid] + s;
        }

        // --- 5: rescale accumulators by exp(mOld - mNew) ---
#pragma unroll
        for (int v = 0; v < 8; ++v) {
            float f = sF[rsel + v];
            acc0[v] *= f;
            acc1[v] *= f;
        }

        // --- 6: acc += P(16x32) @ h_tile(32x32cols) via fp32 WMMA chain ---
#pragma unroll
        for (int k0 = 0; k0 < 32; k0 += 4) {
            v2f afrag;                       // A[M=m16][K=k0+koff .. +1] from LDS
            afrag.x = sP[m16][k0 + koff];
            afrag.y = sP[m16][k0 + koff + 1];
            const float* hb = h + (size_t)(j0 + k0 + koff) * OUT_FEAT;
            v2f b0, b1;                      // B[K][N] from h (L2-resident, 8 MB)
            b0.x = hb[cbase + m16];
            b0.y = hb[OUT_FEAT + cbase + m16];
            b1.x = hb[cbase + 16 + m16];
            b1.y = hb[OUT_FEAT + cbase + 16 + m16];
            acc0 = __builtin_amdgcn_wmma_f32_16x16x4_f32(false, afrag, false, b0,
                                                         (short)0, acc0, false, false);
            acc1 = __builtin_amdgcn_wmma_f32_16x16x4_f32(false, afrag, false, b1,
                                                         (short)0, acc1, false, false);
        }
        __syncthreads();   // protect sLogit/sP/sF for next tile
    }

    // --- epilogue: divide by softmax sum, leaky_relu, store ---
#pragma unroll
    for (int v = 0; v < 8; ++v) {
        int lr = rsel + v;
        float inv = 1.0f / sL[lr];
        float o0 = acc0[v] * inv;
        float o1 = acc1[v] * inv;
        o0 = o0 > 0.f ? o0 : ALPHA * o0;
        o1 = o1 > 0.f ? o1 : ALPHA * o1;
        out[(size_t)(r0 + lr) * OUT_FEAT + cbase + m16]      = o0;
        out[(size_t)(r0 + lr) * OUT_FEAT + cbase + 16 + m16] = o1;
    }
}

// ---------------------------------------------------------------------------
extern "C" void kernel_launch(void* const* d_in, const int* in_sizes, int n_in,
                              void* d_out, int out_size, void* d_ws, size_t ws_size,
                              hipStream_t stream) {
    (void)in_sizes; (void)n_in; (void)out_size; (void)ws_size;
    const float* x   = (const float*)d_in[0];
    const int*   adj = (const int*)d_in[1];
    const float* W   = (const float*)d_in[2];
    const float* a   = (const float*)d_in[3];
    float*       out = (float*)d_out;

    float* h   = (float*)d_ws;                              // 8192*256 f32 = 8 MB
    float* src = h + (size_t)N_NODES * OUT_FEAT;            // 8192 f32
    float* dst = src + N_NODES;                             // 8192 f32

    dim3 g1(OUT_FEAT / 16, N_NODES / 16);
    gemm_h_kernel<<<g1, 32, 0, stream>>>(x, W, h);
    attn_coeff_kernel<<<N_NODES / 8, 256, 0, stream>>>(h, a, src, dst);
    gat_attend_kernel<<<N_NODES / 16, 256, 0, stream>>>(h, adj, src, dst, out);
}